// graphSmoothing_9844065042901
// MI455X (gfx1250) — compile-verified
//
#include <hip/hip_runtime.h>

// Graph mean-aggregation, 3 steps. N=100000 nodes, E=1e6 edges, D=64 feats.
// Output layout [N, 4, 64] f32: slice 0 = x, slices 1..3 = successive means.
// Strategy: build CSR (dst -> list of src) once per call in d_ws, then one
// wave32 per dst node gathers & averages its in-neighbor rows (no f32 atomics).
// Hot-loop gathers are staged through LDS with async global->LDS copies:
// one b128 async instruction fetches TWO 256B src rows (per-lane addressing,
// lanes 0-15 -> row j, lanes 16-31 -> row j+1), double-buffered and
// ASYNCcnt-synchronized, hiding the dependent-gather latency.

#define DFEAT     64
#define ROWSTRIDE 256   // (STEPS+1)*DFEAT floats per node in d_out
#define WAVES_PER_BLOCK 8

// Builtin signatures (probe-confirmed for b64 via clang diagnostic):
//   (vNi addrspace(1)* gsrc, vNi addrspace(3)* ldst, imm offset, imm cpol)
typedef int v2i_t __attribute__((vector_size(2 * sizeof(int))));
typedef int v4i_t __attribute__((vector_size(4 * sizeof(int))));
typedef __attribute__((address_space(1))) v2i_t* as1_v2i;
typedef __attribute__((address_space(3))) v2i_t* as3_v2i;
typedef __attribute__((address_space(1))) v4i_t* as1_v4i;
typedef __attribute__((address_space(3))) v4i_t* as3_v4i;

#define HAVE_ASYNC_B64 0
#define HAVE_ASYNC_B128 0
#if defined(__has_builtin)
#if __has_builtin(__builtin_amdgcn_global_load_async_to_lds_b64)
#undef HAVE_ASYNC_B64
#define HAVE_ASYNC_B64 1
#endif
#if __has_builtin(__builtin_amdgcn_global_load_async_to_lds_b128)
#undef HAVE_ASYNC_B128
#define HAVE_ASYNC_B128 1
#endif
#endif

template <int N>
__device__ __forceinline__ void wait_asynccnt_le() {
#if defined(__has_builtin) && __has_builtin(__builtin_amdgcn_s_wait_asynccnt)
  __builtin_amdgcn_s_wait_asynccnt(N);
#else
  asm volatile("s_wait_asynccnt %0" ::"i"(N) : "memory");
#endif
  asm volatile("" ::: "memory");  // compiler fence: LDS was written behind its back
}

// ------------------------------------------------------- gather-mean (hot)
// One wave32 per dst node. Lane l owns features 2l, 2l+1 of the output row.

__global__ __launch_bounds__(256) void gather_mean_kernel(
    const float* __restrict__ prev, float* __restrict__ cur,
    const int* __restrict__ offs, const int* __restrict__ counts,
    const int* __restrict__ csr_src, int n_nodes) {
#if HAVE_ASYNC_B128
  __shared__ __align__(16) float lbuf[WAVES_PER_BLOCK][2][2 * DFEAT];
#elif HAVE_ASYNC_B64
  __shared__ __align__(16) float lbuf[WAVES_PER_BLOCK][2][DFEAT];
#endif
  const int lane = threadIdx.x & 31;
  const int w = threadIdx.x >> 5;
  const int node = blockIdx.x * WAVES_PER_BLOCK + w;
  if (node >= n_nodes) return;

  const int beg = offs[node];
  const int k = counts[node];
  const int fo = lane * 2;
  float a0 = 0.f, a1 = 0.f;

#if HAVE_ASYNC_B128
  // Two src rows per async b128 issue: lanes 0-15 cover row j (16B each),
  // lanes 16-31 cover row j+1. 512B per instruction, one row pair per wait.
  const int kp = k & ~1;
  const int half = lane >> 4;        // 0: row j, 1: row j+1
  const int q16 = (lane & 15) * 4;   // 16B chunk within the row
  if (kp > 0) {
    int s0 = csr_src[beg];
    int s1 = csr_src[beg + 1];
    const float* g = prev + (size_t)(half ? s1 : s0) * ROWSTRIDE + q16;
    __builtin_amdgcn_global_load_async_to_lds_b128(
        (as1_v4i)(void*)const_cast<float*>(g),
        (as3_v4i)(void*)&lbuf[w][0][lane * 4], 0, 0);
  }
  for (int jp = 0; jp < kp; jp += 2) {
    const int b = (jp >> 1) & 1;
    if (jp + 2 < kp) {
      int s0 = csr_src[beg + jp + 2];
      int s1 = csr_src[beg + jp + 3];
      const float* g = prev + (size_t)(half ? s1 : s0) * ROWSTRIDE + q16;
      // make sure our previous reads of buffer b^1 fully retired
      asm volatile("s_wait_dscnt 0x0" ::: "memory");
      __builtin_amdgcn_global_load_async_to_lds_b128(
          (as1_v4i)(void*)const_cast<float*>(g),
          (as3_v4i)(void*)&lbuf[w][b ^ 1][lane * 4], 0, 0);
      wait_asynccnt_le<1>();  // buffer b complete
    } else {
      wait_asynccnt_le<0>();
    }
    a0 += lbuf[w][b][fo] + lbuf[w][b][DFEAT + fo];
    a1 += lbuf[w][b][fo + 1] + lbuf[w][b][DFEAT + fo + 1];
  }
  if (k & 1) {  // odd-degree tail: one direct 256B row read
    const float* r = prev + (size_t)csr_src[beg + k - 1] * ROWSTRIDE + fo;
    a0 += r[0];
    a1 += r[1];
  }
#elif HAVE_ASYNC_B64
  if (k > 0) {
    {
      int s0 = csr_src[beg];
      __builtin_amdgcn_global_load_async_to_lds_b64(
          (as1_v2i)(void*)const_cast<float*>(prev + (size_t)s0 * ROWSTRIDE + fo),
          (as3_v2i)(void*)&lbuf[w][0][fo], 0, 0);
    }
    for (int j = 0; j < k; ++j) {
      if (j + 1 < k) {
        int sn = csr_src[beg + j + 1];
        asm volatile("s_wait_dscnt 0x0" ::: "memory");
        __builtin_amdgcn_global_load_async_to_lds_b64(
            (as1_v2i)(void*)const_cast<float*>(prev + (size_t)sn * ROWSTRIDE + fo),
            (as3_v2i)(void*)&lbuf[w][(j + 1) & 1][fo], 0, 0);
        wait_asynccnt_le<1>();
      } else {
        wait_asynccnt_le<0>();
      }
      a0 += lbuf[w][j & 1][fo];
      a1 += lbuf[w][j & 1][fo + 1];
    }
  }
#else
  for (int j = 0; j < k; ++j) {
    const float* r = prev + (size_t)csr_src[beg + j] * ROWSTRIDE + fo;
    a0 += r[0];
    a1 += r[1];
  }
#endif

  const float inv = 1.0f / (float)(k > 0 ? k : 1);  // deg.clamp(min=1)
  float2 o;
  o.x = a0 * inv;
  o.y = a1 * inv;
  *(float2*)(cur + (size_t)node * ROWSTRIDE + fo) = o;
}

// ---------------------------------------------------------------- CSR build

__global__ void zero_counts_kernel(int* __restrict__ counts, int n) {
  int i = blockIdx.x * blockDim.x + threadIdx.x;
  if (i < n) counts[i] = 0;
}

__global__ void count_deg_kernel(const long long* __restrict__ ei, int ne,
                                 int* __restrict__ counts) {
  int e = blockIdx.x * blockDim.x + threadIdx.x;
  if (e < ne) atomicAdd(&counts[(int)ei[(size_t)ne + e]], 1);
}

// Single-block exclusive scan over ~100K bins (runs once per call; cheap).
__global__ __launch_bounds__(1024) void scan_kernel(const int* __restrict__ counts,
                                                    int* __restrict__ offs,
                                                    int* __restrict__ cursor, int n) {
  __shared__ int ssum[1024];
  const int t = threadIdx.x;
  const int chunk = (n + 1023) >> 10;
  const int beg = t * chunk;
  const int end = (beg + chunk < n) ? (beg + chunk) : n;
  int s = 0;
  for (int i = beg; i < end; ++i) s += counts[i];
  ssum[t] = s;
  __syncthreads();
  if (t == 0) {  // serial scan of 1024 partials: negligible, once per call
    int run = 0;
    for (int i = 0; i < 1024; ++i) { int v = ssum[i]; ssum[i] = run; run += v; }
  }
  __syncthreads();
  int run = ssum[t];
  for (int i = beg; i < end; ++i) {
    offs[i] = run;
    cursor[i] = run;
    run += counts[i];
  }
}

__global__ void fill_csr_kernel(const long long* __restrict__ ei, int ne,
                                int* __restrict__ cursor, int* __restrict__ csr_src) {
  int e = blockIdx.x * blockDim.x + threadIdx.x;
  if (e < ne) {
    int dst = (int)ei[(size_t)ne + e];
    int pos = atomicAdd(&cursor[dst], 1);
    csr_src[pos] = (int)ei[e];
  }
}

// ------------------------------------------------------------- slice 0 copy

__global__ void copy_x_kernel(const float* __restrict__ x, float* __restrict__ out,
                              int n) {
  int t = blockIdx.x * blockDim.x + threadIdx.x;  // n*16 threads, float4 each
  if (t < n * 16) {
    int node = t >> 4, q = t & 15;
    float4 v = *(const float4*)(x + (size_t)node * DFEAT + q * 4);
    *(float4*)(out + (size_t)node * ROWSTRIDE + q * 4) = v;
  }
}

// ----------------------------------------------------------------- launcher

extern "C" void kernel_launch(void* const* d_in, const int* in_sizes, int n_in,
                              void* d_out, int out_size, void* d_ws, size_t ws_size,
                              hipStream_t stream) {
  const float* x = (const float*)d_in[0];
  const long long* ei = (const long long*)d_in[1];  // int64 [2, E]
  const int n = in_sizes[0] / DFEAT;
  const int ne = in_sizes[1] / 2;
  float* out = (float*)d_out;

  // workspace: counts[n] | offs[n] | cursor[n] | csr_src[ne]  (~5.2 MB)
  int* counts = (int*)d_ws;
  int* offs = counts + n;
  int* cursor = offs + n;
  int* csr_src = cursor + n;

  const int T = 256;
  zero_counts_kernel<<<(n + T - 1) / T, T, 0, stream>>>(counts, n);
  count_deg_kernel<<<(ne + T - 1) / T, T, 0, stream>>>(ei, ne, counts);
  scan_kernel<<<1, 1024, 0, stream>>>(counts, offs, cursor, n);
  fill_csr_kernel<<<(ne + T - 1) / T, T, 0, stream>>>(ei, ne, cursor, csr_src);
  copy_x_kernel<<<(n * 16 + T - 1) / T, T, 0, stream>>>(x, out, n);

  for (int s = 1; s <= 3; ++s) {
    gather_mean_kernel<<<(n + WAVES_PER_BLOCK - 1) / WAVES_PER_BLOCK, T, 0, stream>>>(
        out + (size_t)(s - 1) * DFEAT, out + (size_t)s * DFEAT, offs, counts, csr_src, n);
  }
}